// FundusQuantumLayer_74053826118341
// MI455X (gfx1250) — compile-verified
//
#include <hip/hip_runtime.h>

typedef __attribute__((ext_vector_type(2))) float v2f;
typedef __attribute__((ext_vector_type(8))) float v8f;

#define NQ 4
#define NL 3

// ds_swizzle XOR pattern (group-of-32): offset = xor_mask<<10 | and_mask(0x1f)
template <int OFF>
__device__ __forceinline__ float swz_xor(float v) {
    return __int_as_float(__builtin_amdgcn_ds_swizzle(__float_as_int(v), OFF));
}

// ---------------------------------------------------------------------------
// Prep: collapse the 3 variational layers (shared weights) into U[16x16]
// (complex), then lay out B = [Re(U)^T | Im(U)^T] in the WMMA f32 16x16x4
// B-operand format: for K-step t (features i = 4t..4t+3), lane l supplies
// B[k=2*(l>>4)+v, n=l&15] in VGPR v  ->  ws[((h*4+t)*32 + l)*2 + v],
// h = 0 (Re) / 1 (Im).
// ---------------------------------------------------------------------------
__global__ __launch_bounds__(32) void qc_prep(const float* __restrict__ w,
                                              float* __restrict__ ws)
{
    __shared__ float Ure[256];
    __shared__ float Uim[256];
    const int t = threadIdx.x;
    if (t < 16) {
        float re[16], im[16];
#pragma unroll
        for (int k = 0; k < 16; ++k) { re[k] = (k == t) ? 1.f : 0.f; im[k] = 0.f; }
        for (int l = 0; l < NL; ++l) {
            for (int q = 0; q < NQ; ++q) {
                const int mq = 8 >> q;                 // qubit0 = MSB
                float c, s, cz, sz;
                __sincosf(0.5f * w[(l*NQ + q)*2 + 0], &s, &c);
                for (int k = 0; k < 16; ++k) if (!(k & mq)) {
                    int k1 = k | mq;
                    float r0 = re[k], r1 = re[k1], i0 = im[k], i1 = im[k1];
                    re[k]  = c*r0 - s*r1;  re[k1] = s*r0 + c*r1;   // RY
                    im[k]  = c*i0 - s*i1;  im[k1] = s*i0 + c*i1;
                }
                __sincosf(0.5f * w[(l*NQ + q)*2 + 1], &sz, &cz);
                for (int k = 0; k < 16; ++k) {          // RZ = diag(e^-ip/2, e^+ip/2)
                    float r = re[k], i = im[k];
                    if (!(k & mq)) { re[k] = r*cz + i*sz; im[k] = i*cz - r*sz; }
                    else           { re[k] = r*cz - i*sz; im[k] = i*cz + r*sz; }
                }
            }
            for (int e = 0; e < NQ; ++e) {              // CNOT ring (0,1)(1,2)(2,3)(3,0)
                int mc = 8 >> e, mt = 8 >> ((e + 1) & 3);
                for (int k = 0; k < 16; ++k) if ((k & mc) && !(k & mt)) {
                    int k1 = k | mt;
                    float tr = re[k]; re[k] = re[k1]; re[k1] = tr;
                    float ti = im[k]; im[k] = im[k1]; im[k1] = ti;
                }
            }
        }
        for (int k = 0; k < 16; ++k) { Ure[k*16 + t] = re[k]; Uim[k*16 + t] = im[k]; }
    }
    __syncthreads();
    const int lane = t, n = lane & 15, hh = lane >> 4;
    for (int h = 0; h < 2; ++h)
        for (int tt = 0; tt < 4; ++tt)
            for (int v = 0; v < 2; ++v) {
                int i = 4*tt + 2*hh + v;                // feature (contraction) index
                float val = h ? Uim[n*16 + i] : Ure[n*16 + i];  // B[i,n] = U_part[n,i]
                ws[((h*4 + tt)*32 + lane)*2 + v] = val;
            }
}

// ---------------------------------------------------------------------------
// Main: one wave = one tile of 16 samples per iteration.
//   Phi[16x32] = Psi[16x16] * W[16x32]  via 8x V_WMMA_F32_16X16X4_F32
//   probs -> <Z_q> via signed ds_swizzle butterfly (Walsh-Hadamard over lanes).
// ---------------------------------------------------------------------------
__global__ __launch_bounds__(256) void qc_main(const float4* __restrict__ x4,
                                               const float* __restrict__ ws,
                                               float* __restrict__ out,
                                               int ntiles, int batch)
{
    const int lane = threadIdx.x & 31;
    const int wave = blockIdx.x * (blockDim.x >> 5) + (threadIdx.x >> 5);
    const int nwav = gridDim.x * (blockDim.x >> 5);
    const int h    = lane >> 4;          // which K-pair this lane supplies
    const int sl   = lane & 15;          // sample row within tile / basis column

    const v2f* wsb = (const v2f*)ws;
    v2f bre[4], bim[4];
#pragma unroll
    for (int t = 0; t < 4; ++t) {
        bre[t] = wsb[(t    )*32 + lane];
        bim[t] = wsb[(4 + t)*32 + lane];
    }

    for (int tile = wave; tile < ntiles; tile += nwav) {
        int s = tile*16 + sl;
        s = (s < batch) ? s : (batch - 1);       // clamp, keep EXEC all-ones
        const float4 xv = x4[s];
        float c0, s0, c1, s1, c2, s2, c3, s3;
        __sincosf(0.5f*xv.x, &s0, &c0);          // |x|~N(0,1): hw sin/cos path is fine
        __sincosf(0.5f*xv.y, &s1, &c1);
        __sincosf(0.5f*xv.z, &s2, &c2);
        __sincosf(0.5f*xv.w, &s3, &c3);
        // psi_j = f0[b0]*f1[b1]*f2[b2]*f3[b3], j = 8b0+4b1+2b2+b3.
        // A layout (16x4 f32): lane half h supplies K = {2h, 2h+1} => b2 = h.
        const float f2  = h ? s2 : c2;
        const float pe0 = f2 * c3, pe1 = f2 * s3;   // b3 = 0 / 1

        v8f dre = {}, dim_ = {};
#pragma unroll
        for (int t = 0; t < 4; ++t) {               // K-step: features 4t..4t+3
            float f01 = ((t & 2) ? s0 : c0) * ((t & 1) ? s1 : c1);
            v2f a; a.x = f01 * pe0; a.y = f01 * pe1;
            dre  = __builtin_amdgcn_wmma_f32_16x16x4_f32(false, a, false, bre[t],
                                                         (short)0, dre,  false, false);
            dim_ = __builtin_amdgcn_wmma_f32_16x16x4_f32(false, a, false, bim[t],
                                                         (short)0, dim_, false, false);
        }

        // D layout: lane sl = basis index n, VGPR r -> sample m = r + 8h.
        // <Z_q> = WHT_k(probs) at single-bit mask k = 8>>q: 4-step butterfly
        // using ds_swizzle SWAPX1/2/4/8 patterns (xor within 16-lane halves).
        float wv[8];
#pragma unroll
        for (int r = 0; r < 8; ++r) {
            float p = dre[r]*dre[r] + dim_[r]*dim_[r];
            float o;
            o = swz_xor<(1 << 10) | 0x1f>(p); p = (lane & 1) ? (o - p) : (p + o);
            o = swz_xor<(2 << 10) | 0x1f>(p); p = (lane & 2) ? (o - p) : (p + o);
            o = swz_xor<(4 << 10) | 0x1f>(p); p = (lane & 4) ? (o - p) : (p + o);
            o = swz_xor<(8 << 10) | 0x1f>(p); p = (lane & 8) ? (o - p) : (p + o);
            wv[r] = p;
        }

        const int k = sl;
        if (k == 8 || k == 4 || k == 2 || k == 1) {
            const int q = (k == 8) ? 0 : (k == 4) ? 1 : (k == 2) ? 2 : 3;
            const int mbase = tile*16 + h*8;
#pragma unroll
            for (int r = 0; r < 8; ++r) {
                int sm = mbase + r;
                if (sm < batch) out[sm*4 + q] = wv[r];
            }
        }
    }
}

extern "C" void kernel_launch(void* const* d_in, const int* in_sizes, int n_in,
                              void* d_out, int out_size, void* d_ws, size_t ws_size,
                              hipStream_t stream)
{
    (void)n_in; (void)out_size; (void)ws_size;
    const float* x = (const float*)d_in[0];      // [B,4] f32
    const float* w = (const float*)d_in[1];      // [3,4,2] f32
    float* out = (float*)d_out;                  // [B,4] f32
    const int batch  = in_sizes[0] / 4;
    const int ntiles = (batch + 15) / 16;

    qc_prep<<<1, 32, 0, stream>>>(w, (float*)d_ws);
    qc_main<<<1024, 256, 0, stream>>>((const float4*)x, (const float*)d_ws,
                                      out, ntiles, batch);
}